// EfficientMemoryGELU_77558519431709
// MI455X (gfx1250) — compile-verified
//
#include <hip/hip_runtime.h>
#include <math.h>

// ---------------------------------------------------------------------------
// EfficientMemoryGELU for MI455X (gfx1250): exact GELU + activation
// compression (outliers + int8 fake-quant sub-outlier columns + rank-16
// randomized low-rank approx via CholeskyQR). All rank-16 GEMMs use
// V_WMMA_F32_16X16X4_F32 (K-loop step 4, M=N=16 tiles, wave32, EXEC all-1s).
// ---------------------------------------------------------------------------

typedef __attribute__((ext_vector_type(2))) float v2f;
typedef __attribute__((ext_vector_type(8))) float v8f;

#define NROWS 16384
#define DCOLS 4096
#define TOTAL_ELEMS ((size_t)NROWS * (size_t)DCOLS)
#define RANK 16
#define KSUB 409            // int(4096 * 0.1)
#define NBINS 8192
#define HIST_RANGE 8.0f
#define ROWS_PER_CN 64

// ---- workspace layout (bytes); total ~2.63 MB ----
#define WS_HIST     0          // 8192 u32
#define WS_COLNORM  32768      // 4096 f32
#define WS_MASK     49152      // 4096 i32
#define WS_SCALARS  65536      // f32: [0]=thresh [1]=scale ; u32 view: [2]=minkey [3]=maxkey
#define WS_OMEGA    65792      // 4096*16 f32
#define WS_B        327936     // 16384*16 f32
#define WS_G        1376512    // 16*16 f32
#define WS_W        1377536    // 16*16 f32 (L^-T)
#define WS_Q        1378560    // 16384*16 f32
#define WS_C        2427136    // 16*4096 f32

// ---------------------------------------------------------------------------
// helpers
// ---------------------------------------------------------------------------
__device__ __forceinline__ v8f wmma_f32_16x16x4(v2f a, v2f b, v8f c) {
  // D(16x16,f32) = A(16x4,f32) * B(4x16,f32) + C
  return __builtin_amdgcn_wmma_f32_16x16x4_f32(
      /*neg_a=*/false, a, /*neg_b=*/false, b,
      /*c_mod=*/(short)0, c, /*reuse_a=*/false, /*reuse_b=*/false);
}

__device__ __forceinline__ unsigned fkey(float f) {  // order-preserving float->uint
  unsigned u = __float_as_uint(f);
  return (u & 0x80000000u) ? ~u : (u | 0x80000000u);
}
__device__ __forceinline__ float unfkey(unsigned k) {
  return (k & 0x80000000u) ? __uint_as_float(k & 0x7FFFFFFFu) : __uint_as_float(~k);
}
__device__ __forceinline__ unsigned hashu(unsigned x) {
  x ^= x >> 16; x *= 0x7feb352dU; x ^= x >> 15; x *= 0x846ca68bU; x ^= x >> 16;
  return x;
}
// remainder r = x with outliers zeroed and sub-outlier columns zeroed
__device__ __forceinline__ float rval(const float* __restrict__ x, int row, int col,
                                      float thr, const int* __restrict__ mask) {
  float v = x[(size_t)row * DCOLS + col];
  if (fabsf(v) > thr) v = 0.0f;
  if (mask[col])      v = 0.0f;
  return v;
}

// ---------------------------------------------------------------------------
// 0) workspace init (deterministic each call)
// ---------------------------------------------------------------------------
__global__ void init_kernel(unsigned* __restrict__ hist, float* __restrict__ colnorm,
                            float* __restrict__ G, unsigned* __restrict__ mm) {
  int i = blockIdx.x * blockDim.x + threadIdx.x;
  if (i < NBINS) hist[i] = 0u;
  if (i < DCOLS) colnorm[i] = 0.0f;
  if (i < 256)   G[i] = 0.0f;
  if (i == 0) { mm[0] = fkey(0.0f); mm[1] = fkey(0.0f); }  // x_sub contains zeros
}

// ---------------------------------------------------------------------------
// 1) deterministic Gaussian projection matrix omega[DCOLS][RANK] (Box-Muller)
// ---------------------------------------------------------------------------
__global__ void omega_kernel(float* __restrict__ omega) {
  int i = blockIdx.x * blockDim.x + threadIdx.x;
  if (i >= DCOLS * RANK) return;
  unsigned h1 = hashu(0x9E3779B9u ^ (unsigned)i);
  unsigned h2 = hashu(0x85EBCA6Bu + (unsigned)i * 2654435761u);
  float u1 = ((float)h1 + 1.0f) * (1.0f / 4294967296.0f);
  float u2 = (float)h2 * (1.0f / 4294967296.0f);
  omega[i] = sqrtf(-2.0f * logf(u1)) * cosf(6.28318530718f * u2);
}

// ---------------------------------------------------------------------------
// 2) |x| histogram (LDS-local, one flush per bin per block)
// ---------------------------------------------------------------------------
__global__ void hist_kernel(const float* __restrict__ x, unsigned* __restrict__ hist) {
  __shared__ unsigned sh[NBINS];
  for (int i = threadIdx.x; i < NBINS; i += blockDim.x) sh[i] = 0u;
  __syncthreads();
  const float s = (float)NBINS / HIST_RANGE;
  const size_t nvec = TOTAL_ELEMS / 4;
  const float4* xv = (const float4*)x;
  const size_t stride = (size_t)gridDim.x * blockDim.x;
  for (size_t i = (size_t)blockIdx.x * blockDim.x + threadIdx.x; i < nvec; i += stride) {
    float4 v = xv[i];
    int b0 = min(NBINS - 1, (int)(fabsf(v.x) * s));
    int b1 = min(NBINS - 1, (int)(fabsf(v.y) * s));
    int b2 = min(NBINS - 1, (int)(fabsf(v.z) * s));
    int b3 = min(NBINS - 1, (int)(fabsf(v.w) * s));
    atomicAdd(&sh[b0], 1u); atomicAdd(&sh[b1], 1u);
    atomicAdd(&sh[b2], 1u); atomicAdd(&sh[b3], 1u);
  }
  __syncthreads();
  for (int i = threadIdx.x; i < NBINS; i += blockDim.x)
    if (sh[i]) atomicAdd(&hist[i], sh[i]);
}

// ---------------------------------------------------------------------------
// 3) interpolated 99th percentile of |x|; also writes scalar output
// ---------------------------------------------------------------------------
__global__ void quantile_kernel(const unsigned* __restrict__ hist,
                                float* __restrict__ scalars,
                                float* __restrict__ out_thresh) {
  if (threadIdx.x != 0 || blockIdx.x != 0) return;
  double rank = 0.99 * (double)(TOTAL_ELEMS - 1);
  size_t lo = (size_t)rank;
  double frac = rank - (double)lo;
  size_t cum = 0;
  float thr = HIST_RANGE;
  for (int b = 0; b < NBINS; ++b) {
    size_t c = (size_t)hist[b];
    if (cum + c > lo) {
      double pos = (double)(lo - cum) + frac;
      double den = (c > 0) ? (double)c : 1.0;
      thr = ((float)b + (float)(pos / den)) * (HIST_RANGE / (float)NBINS);
      break;
    }
    cum += c;
  }
  scalars[0] = thr;
  *out_thresh = thr;
}

// ---------------------------------------------------------------------------
// 4) per-column L1 norm of r (outliers removed)
// ---------------------------------------------------------------------------
__global__ void colnorm_kernel(const float* __restrict__ x,
                               const float* __restrict__ scalars,
                               float* __restrict__ colnorm) {
  __shared__ float s[DCOLS];                      // 16 KB of the 320 KB WGP LDS
  const float thr = scalars[0];
  for (int c = threadIdx.x; c < DCOLS; c += blockDim.x) s[c] = 0.0f;
  __syncthreads();
  const int r0 = blockIdx.x * ROWS_PER_CN;
  for (int r = 0; r < ROWS_PER_CN; ++r) {
    const float* xr = x + (size_t)(r0 + r) * DCOLS;
    for (int c = threadIdx.x; c < DCOLS; c += blockDim.x) {
      float v = xr[c];
      float a = fabsf(v);
      s[c] += (a > thr) ? 0.0f : a;               // each thread owns its columns
    }
  }
  __syncthreads();
  for (int c = threadIdx.x; c < DCOLS; c += blockDim.x)
    atomicAdd(&colnorm[c], s[c]);
}

// ---------------------------------------------------------------------------
// 5) top-KSUB columns: binary search on the k-th largest L1 norm -> mask
// ---------------------------------------------------------------------------
__global__ void topk_kernel(const float* __restrict__ colnorm, int* __restrict__ mask) {
  __shared__ float s[DCOLS];
  __shared__ float fred[256];
  __shared__ int   ired[256];
  __shared__ float lohi[2];
  const int tid = threadIdx.x;
  for (int c = tid; c < DCOLS; c += 256) s[c] = colnorm[c];
  __syncthreads();
  float mx = 0.0f;
  for (int c = tid; c < DCOLS; c += 256) mx = fmaxf(mx, s[c]);
  fred[tid] = mx; __syncthreads();
  for (int st = 128; st > 0; st >>= 1) {
    if (tid < st) fred[tid] = fmaxf(fred[tid], fred[tid + st]);
    __syncthreads();
  }
  if (tid == 0) { lohi[0] = 0.0f; lohi[1] = fred[0] + 1.0f; }
  __syncthreads();
  for (int it = 0; it < 40; ++it) {
    float mid = 0.5f * (lohi[0] + lohi[1]);
    int cnt = 0;
    for (int c = tid; c < DCOLS; c += 256) cnt += (s[c] > mid) ? 1 : 0;
    ired[tid] = cnt; __syncthreads();
    for (int st = 128; st > 0; st >>= 1) {
      if (tid < st) ired[tid] += ired[tid + st];
      __syncthreads();
    }
    if (tid == 0) { if (ired[0] >= KSUB) lohi[0] = mid; else lohi[1] = mid; }
    __syncthreads();
  }
  const float t = lohi[0];
  for (int c = tid; c < DCOLS; c += 256) mask[c] = (s[c] > t) ? 1 : 0;
}

// ---------------------------------------------------------------------------
// 6) global min/max of x_sub (masked columns of r), orderable-uint atomics
// ---------------------------------------------------------------------------
__global__ void minmax_kernel(const float* __restrict__ x, const int* __restrict__ mask,
                              const float* __restrict__ scalars, unsigned* __restrict__ mm) {
  const float thr = scalars[0];
  unsigned kmin = 0xFFFFFFFFu, kmax = 0u;
  const size_t stride = (size_t)gridDim.x * blockDim.x;
  for (size_t i = (size_t)blockIdx.x * blockDim.x + threadIdx.x; i < TOTAL_ELEMS; i += stride) {
    const int col = (int)(i & (DCOLS - 1));
    if (!mask[col]) continue;
    float v = x[i];
    if (fabsf(v) > thr) v = 0.0f;
    const unsigned k = fkey(v);
    kmin = (k < kmin) ? k : kmin;
    kmax = (k > kmax) ? k : kmax;
  }
  __shared__ unsigned smin[256], smax[256];
  const int tid = threadIdx.x;
  smin[tid] = kmin; smax[tid] = kmax;
  __syncthreads();
  for (int st = 128; st > 0; st >>= 1) {
    if (tid < st) {
      smin[tid] = (smin[tid + st] < smin[tid]) ? smin[tid + st] : smin[tid];
      smax[tid] = (smax[tid + st] > smax[tid]) ? smax[tid + st] : smax[tid];
    }
    __syncthreads();
  }
  if (tid == 0) { atomicMin(&mm[0], smin[0]); atomicMax(&mm[1], smax[0]); }
}

__global__ void scale_kernel(const unsigned* __restrict__ mm, float* __restrict__ scalars) {
  if (threadIdx.x != 0 || blockIdx.x != 0) return;
  const float mn = unfkey(mm[0]);
  const float mx = unfkey(mm[1]);
  float s = (mx - mn) / 256.0f;                   // 2**SUB_BIT
  scalars[1] = (fabsf(s) > 1e-20f) ? s : 1.0f;
}

// ---------------------------------------------------------------------------
// 7) B = r @ omega   [16384x4096]@[4096x16]  — one wave per 16-row tile
// ---------------------------------------------------------------------------
__global__ void gemmB_kernel(const float* __restrict__ x, const float* __restrict__ omega,
                             const int* __restrict__ mask, const float* __restrict__ scalars,
                             float* __restrict__ B) {
  const int lane = threadIdx.x;
  const int m0 = blockIdx.x * 16;
  const float thr = scalars[0];
  const int m  = lane & 15;
  const int kb = (lane >> 4) << 1;                // 0 for lanes 0-15, 2 for 16-31
  const int row = m0 + m;
  v8f acc = {};
  for (int k = 0; k < DCOLS; k += 4) {
    const int c = k + kb;
    v2f a, b;
    a.x = rval(x, row, c,     thr, mask);         // A[m][k']  = r[row][k+k']
    a.y = rval(x, row, c + 1, thr, mask);
    b.x = omega[(size_t)c * RANK + m];            // B[k'][n]  = omega[k+k'][n], n=lane&15
    b.y = omega[(size_t)(c + 1) * RANK + m];
    acc = wmma_f32_16x16x4(a, b, acc);
  }
  const int rb = (lane < 16) ? 0 : 8;
#pragma unroll
  for (int j = 0; j < 8; ++j)
    B[(size_t)(m0 + rb + j) * RANK + (lane & 15)] = acc[j];
}

// ---------------------------------------------------------------------------
// 8) G = B^T B (16x16) — 64 waves, partial Gram tiles reduced with f32 atomics
//    A[m][k'] = B[r0+k'][m] and Bop[k'][n] = B[r0+k'][n] coincide per lane.
// ---------------------------------------------------------------------------
__global__ void gram_kernel(const float* __restrict__ B, float* __restrict__ G) {
  const int lane = threadIdx.x;
  const int m  = lane & 15;
  const int kb = (lane >> 4) << 1;
  const int n0 = blockIdx.x * (NROWS / 64);
  v8f acc = {};
  for (int nn = 0; nn < NROWS / 64; nn += 4) {
    const int r0 = n0 + nn + kb;
    v2f a;
    a.x = B[(size_t)r0 * RANK + m];
    a.y = B[(size_t)(r0 + 1) * RANK + m];
    acc = wmma_f32_16x16x4(a, a, acc);
  }
  const int rb = (lane < 16) ? 0 : 8;
#pragma unroll
  for (int j = 0; j < 8; ++j)
    atomicAdd(&G[(rb + j) * 16 + (lane & 15)], acc[j]);
}

// ---------------------------------------------------------------------------
// 9) Cholesky of G, W = L^-T (CholeskyQR: Q = B @ W) — single thread, 16x16
// ---------------------------------------------------------------------------
__global__ void chol_kernel(const float* __restrict__ G, float* __restrict__ W) {
  if (threadIdx.x != 0 || blockIdx.x != 0) return;
  float L[16][16], Mi[16][16];
  for (int i = 0; i < 16; ++i)
    for (int j = 0; j < 16; ++j) { L[i][j] = 0.0f; Mi[i][j] = 0.0f; }
  for (int j = 0; j < 16; ++j) {
    float d = G[j * 16 + j];
    for (int p = 0; p < j; ++p) d -= L[j][p] * L[j][p];
    d = sqrtf(fmaxf(d, 1e-20f));
    L[j][j] = d;
    for (int i = j + 1; i < 16; ++i) {
      float s = G[i * 16 + j];
      for (int p = 0; p < j; ++p) s -= L[i][p] * L[j][p];
      L[i][j] = s / d;
    }
  }
  for (int i = 0; i < 16; ++i) {                  // Mi = L^-1 (forward subst)
    Mi[i][i] = 1.0f / L[i][i];
    for (int j = 0; j < i; ++j) {
      float s = 0.0f;
      for (int p = j; p < i; ++p) s += L[i][p] * Mi[p][j];
      Mi[i][j] = -s / L[i][i];
    }
  }
  for (int i = 0; i < 16; ++i)                    // W = (L^-1)^T
    for (int j = 0; j < 16; ++j) W[i * 16 + j] = Mi[j][i];
}

// ---------------------------------------------------------------------------
// 10) Q = B @ W  [16384x16]@[16x16] — one wave per 16-row tile, 4 WMMAs
// ---------------------------------------------------------------------------
__global__ void q_kernel(const float* __restrict__ B, const float* __restrict__ W,
                         float* __restrict__ Q) {
  const int lane = threadIdx.x;
  const int m0 = blockIdx.x * 16;
  const int m  = lane & 15;
  const int kb = (lane >> 4) << 1;
  v8f acc = {};
#pragma unroll
  for (int k = 0; k < RANK; k += 4) {
    v2f a, b;
    a.x = B[(size_t)(m0 + m) * RANK + k + kb];
    a.y = B[(size_t)(m0 + m) * RANK + k + kb + 1];
    b.x = W[(k + kb) * 16 + m];
    b.y = W[(k + kb + 1) * 16 + m];
    acc = wmma_f32_16x16x4(a, b, acc);
  }
  const int rb = (lane < 16) ? 0 : 8;
#pragma unroll
  for (int j = 0; j < 8; ++j)
    Q[(size_t)(m0 + rb + j) * RANK + (lane & 15)] = acc[j];
}

// ---------------------------------------------------------------------------
// 11) C = Q^T @ r  [16x16384]@[16384x4096] — one wave per 16-column tile
// ---------------------------------------------------------------------------
__global__ void c_kernel(const float* __restrict__ x, const float* __restrict__ Q,
                         const int* __restrict__ mask, const float* __restrict__ scalars,
                         float* __restrict__ C) {
  const int lane = threadIdx.x;
  const int c0 = blockIdx.x * 16;
  const float thr = scalars[0];
  const int m  = lane & 15;
  const int kb = (lane >> 4) << 1;
  const int col = c0 + m;
  const int msk = mask[col];
  v8f acc = {};
  for (int n = 0; n < NROWS; n += 4) {
    const int r0 = n + kb;
    __builtin_prefetch(&x[(size_t)(r0 + 16) * DCOLS + col], 0, 0);
    v2f a, b;
    a.x = Q[(size_t)r0 * RANK + m];               // A[m][k'] = Q[n+k'][m]
    a.y = Q[(size_t)(r0 + 1) * RANK + m];
    float v0 = x[(size_t)r0 * DCOLS + col];
    float v1 = x[(size_t)(r0 + 1) * DCOLS + col];
    if (fabsf(v0) > thr) v0 = 0.0f;
    if (fabsf(v1) > thr) v1 = 0.0f;
    b.x = msk ? 0.0f : v0;                        // B[k'][n'] = r[n+k'][c0+n']
    b.y = msk ? 0.0f : v1;
    acc = wmma_f32_16x16x4(a, b, acc);
  }
  const int rb = (lane < 16) ? 0 : 8;
#pragma unroll
  for (int j = 0; j < 8; ++j)
    C[(size_t)(rb + j) * DCOLS + c0 + (lane & 15)] = acc[j];
}

// ---------------------------------------------------------------------------
// 12) fused final pass: r_lr = Q@C per 16x16 tile (4 WMMAs), then
//     x_comp = x_out + x_sub_q + r_lr, and exact erf-GELU — one read of x
//     yields both outputs.
// ---------------------------------------------------------------------------
__global__ void final_kernel(const float* __restrict__ x, const float* __restrict__ Q,
                             const float* __restrict__ C, const int* __restrict__ mask,
                             const float* __restrict__ scalars,
                             float* __restrict__ out_gelu, float* __restrict__ out_comp) {
  const int lane = threadIdx.x;
  const int tile = blockIdx.x;
  const int m0 = (tile >> 8) * 16;                // 1024 row tiles
  const int c0 = (tile & 255) * 16;               // 256 column tiles
  const float thr = scalars[0];
  const float scale = scalars[1];
  const float inv_scale = 1.0f / scale;
  const int m  = lane & 15;
  const int kb = (lane >> 4) << 1;
  v8f acc = {};
#pragma unroll
  for (int k = 0; k < RANK; k += 4) {
    v2f a, b;
    a.x = Q[(size_t)(m0 + m) * RANK + k + kb];
    a.y = Q[(size_t)(m0 + m) * RANK + k + kb + 1];
    b.x = C[(size_t)(k + kb) * DCOLS + c0 + m];
    b.y = C[(size_t)(k + kb + 1) * DCOLS + c0 + m];
    acc = wmma_f32_16x16x4(a, b, acc);
  }
  const int rb = (lane < 16) ? 0 : 8;
  const int col = c0 + (lane & 15);
  const int msk = mask[col];
#pragma unroll
  for (int j = 0; j < 8; ++j) {
    const int row = m0 + rb + j;
    const size_t idx = (size_t)row * DCOLS + col;
    const float v = x[idx];
    out_gelu[idx] = 0.5f * v * (1.0f + erff(v * 0.70710678118654752f));
    const bool is_out = fabsf(v) > thr;
    const float rv = is_out ? 0.0f : v;
    float comp = (is_out ? v : 0.0f) + acc[j];    // x_out + r_lr
    if (msk) comp += rintf(rv * inv_scale) * scale;  // fake int8 quant (RNE)
    out_comp[idx] = comp;
  }
}

// ---------------------------------------------------------------------------
extern "C" void kernel_launch(void* const* d_in, const int* in_sizes, int n_in,
                              void* d_out, int out_size, void* d_ws, size_t ws_size,
                              hipStream_t stream) {
  const float* x = (const float*)d_in[0];
  float* out = (float*)d_out;                     // [gelu | x_comp | thresh]
  char* ws = (char*)d_ws;

  unsigned* hist   = (unsigned*)(ws + WS_HIST);
  float*    cnorm  = (float*)(ws + WS_COLNORM);
  int*      mask   = (int*)(ws + WS_MASK);
  float*    scal   = (float*)(ws + WS_SCALARS);
  unsigned* mm     = (unsigned*)(ws + WS_SCALARS) + 2;
  float*    omega  = (float*)(ws + WS_OMEGA);
  float*    Bm     = (float*)(ws + WS_B);
  float*    G      = (float*)(ws + WS_G);
  float*    W      = (float*)(ws + WS_W);
  float*    Q      = (float*)(ws + WS_Q);
  float*    C      = (float*)(ws + WS_C);

  init_kernel<<<(NBINS + 255) / 256, 256, 0, stream>>>(hist, cnorm, G, mm);
  omega_kernel<<<(DCOLS * RANK + 255) / 256, 256, 0, stream>>>(omega);
  hist_kernel<<<2048, 256, 0, stream>>>(x, hist);
  quantile_kernel<<<1, 1, 0, stream>>>(hist, scal, out + 2 * TOTAL_ELEMS);
  colnorm_kernel<<<NROWS / ROWS_PER_CN, 256, 0, stream>>>(x, scal, cnorm);
  topk_kernel<<<1, 256, 0, stream>>>(cnorm, mask);
  minmax_kernel<<<2048, 256, 0, stream>>>(x, mask, scal, mm);
  scale_kernel<<<1, 1, 0, stream>>>(mm, scal);
  gemmB_kernel<<<NROWS / 16, 32, 0, stream>>>(x, omega, mask, scal, Bm);
  gram_kernel<<<64, 32, 0, stream>>>(Bm, G);
  chol_kernel<<<1, 1, 0, stream>>>(G, W);
  q_kernel<<<NROWS / 16, 32, 0, stream>>>(Bm, W, Q);
  c_kernel<<<DCOLS / 16, 32, 0, stream>>>(x, Q, mask, scal, C);
  final_kernel<<<(NROWS / 16) * (DCOLS / 16), 32, 0, stream>>>(
      x, Q, C, mask, scal, out, out + TOTAL_ELEMS);
}